// ChamferLoss_70068096467887
// MI455X (gfx1250) — compile-verified
//
#include <hip/hip_runtime.h>

typedef __attribute__((ext_vector_type(2))) float v2f;
typedef __attribute__((ext_vector_type(8))) float v8f;

#define BATCH 2
#define NPTS  8192
#define MPTS  2048
#define KNN_K 10
#define KTILE 2048   // KNN LDS tile (24KB)
#define STILE 2048   // WMMA column tile (<=40KB LDS)

// ---------------------------------------------------------------------------
// squared norms per point
// ---------------------------------------------------------------------------
__global__ void sqnorm_kernel(const float* __restrict__ p, float* __restrict__ sq, int total) {
    int i = blockIdx.x * blockDim.x + threadIdx.x;
    if (i < total) {
        float x = p[3 * i + 0], y = p[3 * i + 1], z = p[3 * i + 2];
        sq[i] = x * x + y * y + z * z;
    }
}

// ---------------------------------------------------------------------------
// analytic smallest eigenvalue of a symmetric 3x3 matrix (trig method)
// ---------------------------------------------------------------------------
__device__ inline float smallest_eig3(float sxx, float syy, float szz,
                                      float sxy, float sxz, float syz, float tr) {
    float p1 = sxy * sxy + sxz * sxz + syz * syz;
    float q  = tr * (1.0f / 3.0f);
    float dxx = sxx - q, dyy = syy - q, dzz = szz - q;
    float p2 = dxx * dxx + dyy * dyy + dzz * dzz + 2.0f * p1;
    float p  = sqrtf(fmaxf(p2 * (1.0f / 6.0f), 0.0f));
    if (p < 1e-20f) return q;                       // ~scalar matrix
    float ip = 1.0f / p;
    float bxx = dxx * ip, byy = dyy * ip, bzz = dzz * ip;
    float bxy = sxy * ip, bxz = sxz * ip, byz = syz * ip;
    float detB = bxx * (byy * bzz - byz * byz)
               - bxy * (bxy * bzz - byz * bxz)
               + bxz * (bxy * byz - byy * bxz);
    float r   = fminf(fmaxf(0.5f * detB, -1.0f), 1.0f);
    float phi = acosf(r) * (1.0f / 3.0f);
    return q + 2.0f * p * __cosf(phi + 2.0943951023931953f); // + 2*pi/3 -> smallest
}

// ---------------------------------------------------------------------------
// KNN (k=10) + curvature. One thread per point; register-resident top-k.
// Gram (k x k) of k x 3 diffs is PSD rank<=3 -> its smallest eig = min(lam3,0).
// ---------------------------------------------------------------------------
__global__ void knn_curv_kernel(const float* __restrict__ p1, float* __restrict__ curv,
                                const int* __restrict__ kptr, int N) {
    __shared__ float sx[KTILE], sy[KTILE], sz[KTILE];
    const int b = blockIdx.y;
    const float* P = p1 + (size_t)b * N * 3;
    const int n = blockIdx.x * blockDim.x + threadIdx.x;
    const float px = P[n * 3 + 0], py = P[n * 3 + 1], pz = P[n * 3 + 2];

    float bd[KNN_K];
    int   bi[KNN_K];
#pragma unroll
    for (int t = 0; t < KNN_K; ++t) { bd[t] = 3.0e38f; bi[t] = 0; }

    for (int t0 = 0; t0 < N; t0 += KTILE) {
        __syncthreads();
        for (int i = threadIdx.x; i < KTILE; i += blockDim.x) {
            int m = t0 + i;
            sx[i] = P[m * 3 + 0]; sy[i] = P[m * 3 + 1]; sz[i] = P[m * 3 + 2];
        }
        __syncthreads();
        if (t0 + KTILE < N) __builtin_prefetch(&P[(t0 + KTILE) * 3], 0, 1);
        for (int j = 0; j < KTILE; ++j) {
            float dx = px - sx[j], dy = py - sy[j], dz = pz - sz[j];
            float d2 = dx * dx + dy * dy + dz * dz;
            if (d2 < bd[KNN_K - 1]) {
                float cd = d2; int ci = t0 + j;
#pragma unroll
                for (int t = 0; t < KNN_K; ++t) {
                    if (cd < bd[t]) {
                        float td = bd[t]; bd[t] = cd; cd = td;
                        int   ti = bi[t]; bi[t] = ci; ci = ti;
                    }
                }
            }
        }
    }

    // 3x3 scatter of neighbor diffs
    float sxx = 0, syy = 0, szz = 0, sxy = 0, sxz = 0, syz = 0;
#pragma unroll
    for (int t = 0; t < KNN_K; ++t) {
        int m = bi[t];
        float dx = P[m * 3 + 0] - px, dy = P[m * 3 + 1] - py, dz = P[m * 3 + 2] - pz;
        sxx += dx * dx; syy += dy * dy; szz += dz * dz;
        sxy += dx * dy; sxz += dx * dz; syz += dy * dz;
    }
    float kf   = (float)(*kptr);
    float tr   = sxx + syy + szz;
    float lmin = smallest_eig3(sxx, syy, szz, sxy, sxz, syz, tr);
    float gmin = fminf(lmin, 0.0f) / kf;   // k > 3 => Gram has zero eigenvalues
    float s    = tr / kf;
    curv[(size_t)b * N + n] = (s != 0.0f) ? (gmin / s) : 0.0f;
}

// ---------------------------------------------------------------------------
// WMMA fragment helpers for V_WMMA_F32_16X16X4_F32 (16x4 A / 4x16 B, K padded)
// lanes 0-15:  (x, y) -> K0,K1 ; lanes 16-31: (z, 0) -> K2,K3
// ---------------------------------------------------------------------------
__device__ inline v2f make_frag_global(const float* __restrict__ P, int idx, int half) {
    v2f f;
    if (half == 0) { f.x = P[idx * 3 + 0]; f.y = P[idx * 3 + 1]; }
    else           { f.x = P[idx * 3 + 2]; f.y = 0.0f; }
    return f;
}

__device__ inline v8f wmma_dot(v2f a, v2f b) {
    v8f c = {};
#if __has_builtin(__builtin_amdgcn_wmma_f32_16x16x4_f32)
    c = __builtin_amdgcn_wmma_f32_16x16x4_f32(false, a, false, b, (short)0, c,
                                              false, false);
#else
    (void)a; (void)b;   // host pass only; never executed on device
#endif
    return c;
}

// ---------------------------------------------------------------------------
// Curvature smoothing: smoothed[n] = sum_m curv[m] e^{-10 d2} / sum_m e^{-10 d2}
// One wave per 16-row strip; WMMA produces 16x16 dot-product tiles.
// ---------------------------------------------------------------------------
__global__ void smooth_kernel(const float* __restrict__ p1, const float* __restrict__ sq1,
                              const float* __restrict__ curv, float* __restrict__ outSm, int N) {
    __shared__ float sx[STILE], sy[STILE], sz[STILE], ssq[STILE], scv[STILE];
    const int b = blockIdx.y;
    const float* P  = p1   + (size_t)b * N * 3;
    const float* SQ = sq1  + (size_t)b * N;
    const float* CV = curv + (size_t)b * N;

    const int lane = threadIdx.x & 31;
    const int wave = threadIdx.x >> 5;
    const int n0   = (blockIdx.x * 8 + wave) * 16;
    const int half = lane >> 4;
    const int li   = lane & 15;

    v2f afrag = make_frag_global(P, n0 + li, half);
    float rowaa[8];
#pragma unroll
    for (int v = 0; v < 8; ++v) rowaa[v] = SQ[n0 + v + 8 * half];

    float wsum[8], wcs[8];
#pragma unroll
    for (int v = 0; v < 8; ++v) { wsum[v] = 0.0f; wcs[v] = 0.0f; }

    for (int t0 = 0; t0 < N; t0 += STILE) {
        __syncthreads();
        for (int i = threadIdx.x; i < STILE; i += blockDim.x) {
            int m = t0 + i;
            sx[i] = P[m * 3 + 0]; sy[i] = P[m * 3 + 1]; sz[i] = P[m * 3 + 2];
            ssq[i] = SQ[m]; scv[i] = CV[m];
        }
        __syncthreads();
        for (int c0 = 0; c0 < STILE; c0 += 16) {
            int col = c0 + li;
            v2f bfrag;
            if (half == 0) { bfrag.x = sx[col]; bfrag.y = sy[col]; }
            else           { bfrag.x = sz[col]; bfrag.y = 0.0f; }
            v8f c = wmma_dot(afrag, bfrag);
            float bb = ssq[col];
            float cv = scv[col];
#pragma unroll
            for (int v = 0; v < 8; ++v) {
                float d2 = fmaxf(rowaa[v] + bb - 2.0f * c[v], 0.0f);
                float w  = __expf(-10.0f * d2);     // exp(-d2 / 0.1)
                wsum[v] += w;
                wcs[v]  += w * cv;
            }
        }
    }
    // reduce over the N dimension: 16 lanes per half share a row
#pragma unroll
    for (int v = 0; v < 8; ++v) {
#pragma unroll
        for (int m = 1; m < 16; m <<= 1) {
            wsum[v] += __shfl_xor(wsum[v], m, 16);
            wcs[v]  += __shfl_xor(wcs[v],  m, 16);
        }
    }
    if (li == 0) {
#pragma unroll
        for (int v = 0; v < 8; ++v) {
            int row = n0 + v + 8 * half;
            outSm[(size_t)b * N + row] = wcs[v] / wsum[v];
        }
    }
}

// ---------------------------------------------------------------------------
// Chamfer min over columns for 16-row strips (both directions reuse this).
// Writes one deterministic partial sum per strip.
// ---------------------------------------------------------------------------
__global__ void chamfer_kernel(const float* __restrict__ Prow, const float* __restrict__ SQrow, int NR,
                               const float* __restrict__ Pcol, const float* __restrict__ SQcol, int NC,
                               float* __restrict__ partial) {
    __shared__ float sx[STILE], sy[STILE], sz[STILE], ssq[STILE];
    const int b = blockIdx.y;
    const float* PR = Prow + (size_t)b * NR * 3;
    const float* SR = SQrow + (size_t)b * NR;
    const float* PC = Pcol + (size_t)b * NC * 3;
    const float* SC = SQcol + (size_t)b * NC;

    const int lane = threadIdx.x & 31;
    const int wave = threadIdx.x >> 5;
    const int strip = blockIdx.x * 8 + wave;
    const int n0   = strip * 16;
    const int half = lane >> 4;
    const int li   = lane & 15;

    v2f afrag = make_frag_global(PR, n0 + li, half);
    float rowaa[8];
#pragma unroll
    for (int v = 0; v < 8; ++v) rowaa[v] = SR[n0 + v + 8 * half];

    float minv[8];
#pragma unroll
    for (int v = 0; v < 8; ++v) minv[v] = 3.0e38f;

    for (int t0 = 0; t0 < NC; t0 += STILE) {
        int cnt = (NC - t0 < STILE) ? (NC - t0) : STILE;
        __syncthreads();
        for (int i = threadIdx.x; i < cnt; i += blockDim.x) {
            int m = t0 + i;
            sx[i] = PC[m * 3 + 0]; sy[i] = PC[m * 3 + 1]; sz[i] = PC[m * 3 + 2];
            ssq[i] = SC[m];
        }
        __syncthreads();
        for (int c0 = 0; c0 < cnt; c0 += 16) {
            int col = c0 + li;
            v2f bfrag;
            if (half == 0) { bfrag.x = sx[col]; bfrag.y = sy[col]; }
            else           { bfrag.x = sz[col]; bfrag.y = 0.0f; }
            v8f c = wmma_dot(afrag, bfrag);
            float bb = ssq[col];
#pragma unroll
            for (int v = 0; v < 8; ++v) {
                float d2 = fmaxf(rowaa[v] + bb - 2.0f * c[v], 0.0f);
                minv[v] = fminf(minv[v], d2);
            }
        }
    }
#pragma unroll
    for (int v = 0; v < 8; ++v) {
#pragma unroll
        for (int m = 1; m < 16; m <<= 1)
            minv[v] = fminf(minv[v], __shfl_xor(minv[v], m, 16));
    }
    // per-half row sums of sqrt(min)^2 (mirror reference rounding)
    float rs = 0.0f;
#pragma unroll
    for (int v = 0; v < 8; ++v) {
        float sd = sqrtf(minv[v]);
        rs += sd * sd;
    }
    float r0  = __shfl(rs, 0, 32);
    float r16 = __shfl(rs, 16, 32);
    if (lane == 0) partial[(size_t)b * (NR / 16) + strip] = r0 + r16;
}

// ---------------------------------------------------------------------------
// Deterministic fixed-order final reduction: out[b] = sum partials
// ---------------------------------------------------------------------------
__global__ void finalize_kernel(const float* __restrict__ pn, int nN,
                                const float* __restrict__ pm, int nM,
                                float* __restrict__ out, int B) {
    int b = threadIdx.x;
    if (b < B) {
        float s = 0.0f;
        for (int i = 0; i < nN; ++i) s += pn[b * nN + i];
        for (int i = 0; i < nM; ++i) s += pm[b * nM + i];
        out[b] = s;
    }
}

// ---------------------------------------------------------------------------
extern "C" void kernel_launch(void* const* d_in, const int* in_sizes, int n_in,
                              void* d_out, int out_size, void* d_ws, size_t ws_size,
                              hipStream_t stream) {
    (void)in_sizes; (void)n_in; (void)out_size; (void)ws_size;
    const float* p1   = (const float*)d_in[0];   // (2, 8192, 3) f32
    const float* p2   = (const float*)d_in[1];   // (2, 2048, 3) f32
    const int*   kptr = (const int*)d_in[2];     // scalar k_neighbors
    float* out = (float*)d_out;                  // [0..1]=total_loss, [2..]=smoothed
    float* ws  = (float*)d_ws;

    const int B = BATCH, N = NPTS, M = MPTS;
    float* curv = ws;                    // B*N
    float* sq1  = curv + B * N;          // B*N
    float* sq2  = sq1  + B * N;          // B*M
    float* pn   = sq2  + B * M;          // B*(N/16)
    float* pm   = pn   + B * (N / 16);   // B*(M/16)

    sqnorm_kernel<<<(B * N + 255) / 256, 256, 0, stream>>>(p1, sq1, B * N);
    sqnorm_kernel<<<(B * M + 255) / 256, 256, 0, stream>>>(p2, sq2, B * M);

    dim3 gk(N / 256, B);
    knn_curv_kernel<<<gk, 256, 0, stream>>>(p1, curv, kptr, N);

    dim3 gs(N / 128, B);                 // 8 waves/block, 16 rows/wave
    smooth_kernel<<<gs, 256, 0, stream>>>(p1, sq1, curv, out + B, N);

    dim3 gc1(N / 128, B);
    chamfer_kernel<<<gc1, 256, 0, stream>>>(p1, sq1, N, p2, sq2, M, pn);
    dim3 gc2(M / 128, B);
    chamfer_kernel<<<gc2, 256, 0, stream>>>(p2, sq2, M, p1, sq1, N, pm);

    finalize_kernel<<<1, 64, 0, stream>>>(pn, N / 16, pm, M / 16, out, B);
}